// AttnSeqDecoder_24970939859740
// MI455X (gfx1250) — compile-verified
//
#include <hip/hip_runtime.h>
#include <math.h>

// Problem constants (match reference)
#define B_  64
#define S_  128
#define H_  1024
#define E_  1024
#define V_  32000
#define L_  2

typedef __attribute__((ext_vector_type(2))) float v2f;
typedef __attribute__((ext_vector_type(8))) float v8f;

// ---------------------------------------------------------------------------
// WMMA fp32 GEMM:  out[64, N] = X1[64,K1] @ W1[N,K1]^T (+ X2 @ W2^T) + bias
//   - block = 128 threads = 4 waves; each wave computes a 64x32 output strip
//   - grid.x * 128 == N
//   - optional row gather on X1 (embedding lookup)
//   - optional tanh activation on store
// Each lane loads float4 at column (k + 16B half-wave offset); the two WMMAs
// per step consume (x,y) then (z,w) with identical slot assignment in A and B,
// so K ordering is consistent between operands (WMMA sums all 4 slots).
// ---------------------------------------------------------------------------
__device__ __forceinline__ void gemm_pass(const float* __restrict__ X, int ldx,
                                          const int* __restrict__ gather,
                                          const float* __restrict__ W, int ldw,
                                          int K, int l16, int hi, int n0,
                                          v8f (&acc)[4][2]) {
    const float* xr[4];
#pragma unroll
    for (int mt = 0; mt < 4; ++mt) {
        int r = mt * 16 + l16;
        if (gather) r = gather[r];
        xr[mt] = X + (size_t)r * (size_t)ldx;
    }
    const float* wr[2];
#pragma unroll
    for (int nt = 0; nt < 2; ++nt)
        wr[nt] = W + (size_t)(n0 + nt * 16 + l16) * (size_t)ldw;

    for (int k = 0; k < K; k += 8) {
        int ko = k + hi * 4;                       // half-wave K offset
        float4 a[4];
#pragma unroll
        for (int mt = 0; mt < 4; ++mt) a[mt] = *(const float4*)(xr[mt] + ko);
        float4 bf[2];
#pragma unroll
        for (int nt = 0; nt < 2; ++nt) bf[nt] = *(const float4*)(wr[nt] + ko);

#pragma unroll
        for (int nt = 0; nt < 2; ++nt) {
            v2f b0 = {bf[nt].x, bf[nt].y};
            v2f b1 = {bf[nt].z, bf[nt].w};
#pragma unroll
            for (int mt = 0; mt < 4; ++mt) {
                v2f a0 = {a[mt].x, a[mt].y};
                v2f a1 = {a[mt].z, a[mt].w};
                acc[mt][nt] = __builtin_amdgcn_wmma_f32_16x16x4_f32(
                    false, a0, false, b0, (short)0, acc[mt][nt], false, false);
                acc[mt][nt] = __builtin_amdgcn_wmma_f32_16x16x4_f32(
                    false, a1, false, b1, (short)0, acc[mt][nt], false, false);
            }
        }
    }
}

__global__ void __launch_bounds__(128) wmma_gemm_xt(
    const float* __restrict__ X1, int ldx1, const int* __restrict__ gather,
    const float* __restrict__ W1, int ldw1, int K1,
    const float* __restrict__ X2, int ldx2,
    const float* __restrict__ W2, int ldw2, int K2,
    const float* __restrict__ bias1, const float* __restrict__ bias2,
    float* __restrict__ out, int ldo, int act_tanh) {
    const int lane = threadIdx.x & 31;
    const int wave = threadIdx.x >> 5;
    const int l16  = lane & 15;
    const int hi   = lane >> 4;
    const int n0   = blockIdx.x * 128 + wave * 32;

    v8f acc[4][2] = {};

    gemm_pass(X1, ldx1, gather, W1, ldw1, K1, l16, hi, n0, acc);
    if (X2) gemm_pass(X2, ldx2, nullptr, W2, ldw2, K2, l16, hi, n0, acc);

    // C/D layout: vgpr j, lanes 0-15 -> M=j, N=lane; lanes 16-31 -> M=j+8.
#pragma unroll
    for (int nt = 0; nt < 2; ++nt) {
        int n = n0 + nt * 16 + l16;
        float bv = bias1 ? bias1[n] : 0.0f;
        if (bias2) bv += bias2[n];
#pragma unroll
        for (int mt = 0; mt < 4; ++mt) {
#pragma unroll
            for (int j = 0; j < 8; ++j) {
                int m = mt * 16 + hi * 8 + j;
                float v = acc[mt][nt][j] + bv;
                if (act_tanh) v = tanhf(v);
                out[(size_t)m * (size_t)ldo + n] = v;
            }
        }
    }
}

// ---------------------------------------------------------------------------
// LSTM elementwise: gates[B,4H] (i,f,g,o) + c0 -> h, c
// ---------------------------------------------------------------------------
__device__ __forceinline__ float sigm(float x) { return 1.0f / (1.0f + expf(-x)); }

__global__ void lstm_pointwise(const float* __restrict__ gates,
                               const float* __restrict__ c0,
                               float* __restrict__ h_out,
                               float* __restrict__ c_out) {
    int idx = blockIdx.x * blockDim.x + threadIdx.x;      // 0..B*H
    int b = idx / H_;
    int j = idx - b * H_;
    const float* g = gates + (size_t)b * (4 * H_);
    float gi = g[j];
    float gf = g[H_ + j];
    float gg = g[2 * H_ + j];
    float go = g[3 * H_ + j];
    float c  = sigm(gf) * c0[idx] + sigm(gi) * tanhf(gg);
    float h  = sigm(go) * tanhf(c);
    h_out[idx] = h;
    c_out[idx] = c;
}

// ---------------------------------------------------------------------------
// Dot attention: one block per batch row. scores -> softmax -> context.
// ---------------------------------------------------------------------------
__global__ void __launch_bounds__(256) attention_k(const float* __restrict__ enc,
                                                   const float* __restrict__ rnn,
                                                   float* __restrict__ attn_out,
                                                   float* __restrict__ ctx_out) {
    const int b = blockIdx.x;
    const int tid = threadIdx.x;
    __shared__ float r[H_];
    __shared__ float sc[S_];
    __shared__ float red[2];

    for (int h = tid; h < H_; h += 256) r[h] = rnn[(size_t)b * H_ + h];
    __syncthreads();

    for (int s = tid; s < S_; s += 256) {
        const float* e = enc + ((size_t)b * S_ + s) * H_;
        float acc = 0.0f;
        for (int h = 0; h < H_; h += 4) {
            float4 ev = *(const float4*)(e + h);
            acc += ev.x * r[h] + ev.y * r[h + 1] + ev.z * r[h + 2] + ev.w * r[h + 3];
        }
        sc[s] = acc;
    }
    __syncthreads();

    if (tid == 0) {
        float mx = -3.4e38f;
        for (int s = 0; s < S_; ++s) mx = fmaxf(mx, sc[s]);
        float sum = 0.0f;
        for (int s = 0; s < S_; ++s) sum += expf(sc[s] - mx);
        red[0] = mx;
        red[1] = sum;
    }
    __syncthreads();

    float mx = red[0];
    float inv = 1.0f / red[1];
    for (int s = tid; s < S_; s += 256) {
        float a = expf(sc[s] - mx) * inv;
        sc[s] = a;
        attn_out[(size_t)b * S_ + s] = a;
    }
    __syncthreads();

    for (int h = tid; h < H_; h += 256) {
        float acc = 0.0f;
        for (int s = 0; s < S_; ++s)
            acc += sc[s] * enc[((size_t)b * S_ + s) * H_ + h];
        ctx_out[(size_t)b * H_ + h] = acc;
    }
}

// ---------------------------------------------------------------------------
// log_softmax over V per row: one block per batch row.
// ---------------------------------------------------------------------------
__global__ void __launch_bounds__(256) log_softmax_k(const float* __restrict__ logits,
                                                     float* __restrict__ out) {
    const int b = blockIdx.x;
    const int tid = threadIdx.x;
    __shared__ float red[256];
    const float* row = logits + (size_t)b * V_;

    float mx = -3.4e38f;
    for (int v = tid; v < V_; v += 256) mx = fmaxf(mx, row[v]);
    red[tid] = mx;
    __syncthreads();
    for (int s = 128; s > 0; s >>= 1) {
        if (tid < s) red[tid] = fmaxf(red[tid], red[tid + s]);
        __syncthreads();
    }
    mx = red[0];
    __syncthreads();

    float sum = 0.0f;
    for (int v = tid; v < V_; v += 256) sum += expf(row[v] - mx);
    red[tid] = sum;
    __syncthreads();
    for (int s = 128; s > 0; s >>= 1) {
        if (tid < s) red[tid] += red[tid + s];
        __syncthreads();
    }
    float lse = mx + logf(red[0]);

    for (int v = tid; v < V_; v += 256)
        out[(size_t)b * V_ + v] = row[v] - lse;
}

// ---------------------------------------------------------------------------
// Host launcher
// ---------------------------------------------------------------------------
extern "C" void kernel_launch(void* const* d_in, const int* in_sizes, int n_in,
                              void* d_out, int out_size, void* d_ws, size_t ws_size,
                              hipStream_t stream) {
    (void)in_sizes; (void)n_in; (void)out_size; (void)ws_size;

    const float* enc  = (const float*)d_in[0];   // [B,S,H]
    const int*   prev = (const int*)  d_in[1];   // [B]
    const float* h0   = (const float*)d_in[2];   // [L,B,H]
    const float* c0   = (const float*)d_in[3];   // [L,B,H]
    const float* emb  = (const float*)d_in[4];   // [V,E]
    const float* wih  = (const float*)d_in[5];   // [L,4H,E]
    const float* whh  = (const float*)d_in[6];   // [L,4H,H]
    const float* bih  = (const float*)d_in[7];   // [L,4H]
    const float* bhh  = (const float*)d_in[8];   // [L,4H]
    const float* mw   = (const float*)d_in[9];   // [H,2H]
    const float* mb   = (const float*)d_in[10];  // [H]
    const float* gw   = (const float*)d_in[11];  // [V,H]
    const float* gb   = (const float*)d_in[12];  // [V]

    float* out       = (float*)d_out;
    float* out_logp  = out;                                  // [B,V]
    float* out_h     = out_logp + (size_t)B_ * V_;           // [L,B,H]
    float* out_c     = out_h + (size_t)L_ * B_ * H_;         // [L,B,H]
    float* out_attn  = out_c + (size_t)L_ * B_ * H_;         // [B,S]

    float* ws        = (float*)d_ws;
    float* ws_gates  = ws;                                   // [B,4H]
    float* ws_ctx    = ws_gates + (size_t)B_ * 4 * H_;       // [B,H]
    float* ws_merged = ws_ctx + (size_t)B_ * H_;             // [B,H]
    float* ws_logits = ws_merged + (size_t)B_ * H_;          // [B,V]

    const dim3 gblk(128);

    // ---- LSTM layer 0: gates = emb[prev] @ w_ih0^T + h0[0] @ w_hh0^T + b ----
    wmma_gemm_xt<<<(4 * H_) / 128, gblk, 0, stream>>>(
        emb, E_, prev, wih, E_, E_,
        h0, H_, whh, H_, H_,
        bih, bhh, ws_gates, 4 * H_, 0);
    lstm_pointwise<<<(B_ * H_) / 256, 256, 0, stream>>>(
        ws_gates, c0, out_h, out_c);

    // ---- LSTM layer 1 ----
    wmma_gemm_xt<<<(4 * H_) / 128, gblk, 0, stream>>>(
        out_h, H_, nullptr, wih + (size_t)4 * H_ * E_, E_, E_,
        h0 + (size_t)B_ * H_, H_, whh + (size_t)4 * H_ * H_, H_, H_,
        bih + 4 * H_, bhh + 4 * H_, ws_gates, 4 * H_, 0);
    lstm_pointwise<<<(B_ * H_) / 256, 256, 0, stream>>>(
        ws_gates, c0 + (size_t)B_ * H_,
        out_h + (size_t)B_ * H_, out_c + (size_t)B_ * H_);

    const float* rnn = out_h + (size_t)B_ * H_;  // top-layer h

    // ---- attention: scores/softmax/context ----
    attention_k<<<B_, 256, 0, stream>>>(enc, rnn, out_attn, ws_ctx);

    // ---- merged = tanh([rnn|ctx] @ merge_w^T + merge_b), split-K halves ----
    wmma_gemm_xt<<<H_ / 128, gblk, 0, stream>>>(
        rnn, H_, nullptr, mw, 2 * H_, H_,
        ws_ctx, H_, mw + H_, 2 * H_, H_,
        mb, nullptr, ws_merged, H_, 1);

    // ---- logits = merged @ gen_w^T + gen_b  (dominant: streams 131 MB once) ----
    wmma_gemm_xt<<<V_ / 128, gblk, 0, stream>>>(
        ws_merged, H_, nullptr, gw, H_, H_,
        nullptr, 0, nullptr, 0, 0,
        gb, nullptr, ws_logits, V_, 0);

    // ---- log_softmax ----
    log_softmax_k<<<B_, 256, 0, stream>>>(ws_logits, out_logp);
}